// HierarchicalEmergence_13984413516426
// MI455X (gfx1250) — compile-verified
//
#include <hip/hip_runtime.h>
#include <math.h>

#define NN 2048
#define MM 128
#define KD 8
#define XC 80   // padded feature row length (72 used: 64 matrix + 8 vector)

typedef float v2f __attribute__((ext_vector_type(2)));
typedef float v8f __attribute__((ext_vector_type(8)));

// ---------- 8x8 helpers (fully unrolled, register resident) ----------

// Gauss-Jordan inverse (no pivoting: Omega ~ I + 0.1*noise, Sigma SPD well-cond).
// Returns log|det|.
static __device__ __forceinline__ float inv8_logdet(const float* __restrict__ in,
                                                    float* __restrict__ out) {
  float a[64];
#pragma unroll
  for (int t = 0; t < 64; t++) a[t] = in[t];
#pragma unroll
  for (int i = 0; i < 8; i++)
#pragma unroll
    for (int j = 0; j < 8; j++) out[i * 8 + j] = (i == j) ? 1.0f : 0.0f;
  float ld = 0.0f;
#pragma unroll
  for (int j = 0; j < 8; j++) {
    float piv = a[j * 8 + j];
    ld += logf(fabsf(piv));
    float ip = 1.0f / piv;
#pragma unroll
    for (int c = 0; c < 8; c++) { a[j * 8 + c] *= ip; out[j * 8 + c] *= ip; }
#pragma unroll
    for (int r = 0; r < 8; r++) {
      if (r == j) continue;
      float f = a[r * 8 + j];
#pragma unroll
      for (int c = 0; c < 8; c++) {
        a[r * 8 + c] -= f * a[j * 8 + c];
        out[r * 8 + c] -= f * out[j * 8 + c];
      }
    }
  }
  return ld;
}

// logdet of SPD 8x8 via unpivoted LU (pivots > 0 for SPD).
static __device__ __forceinline__ float spd_logdet8(const float* __restrict__ in) {
  float a[64];
#pragma unroll
  for (int t = 0; t < 64; t++) a[t] = in[t];
  float ld = 0.0f;
#pragma unroll
  for (int j = 0; j < 8; j++) {
    float piv = a[j * 8 + j];
    ld += logf(piv);
    float ip = 1.0f / piv;
#pragma unroll
    for (int r = j + 1; r < 8; r++) {
      float f = a[r * 8 + j] * ip;
#pragma unroll
      for (int c = j; c < 8; c++) a[r * 8 + c] -= f * a[j * 8 + c];
    }
  }
  return ld;
}

// ---------- pipeline kernels ----------

__global__ void he_init(float* __restrict__ acc) {
  if (threadIdx.x < 2) acc[threadIdx.x] = 0.0f;
}

// Per child i: X_i = [(B_i + u u^T)_flat, -2u, 0pad]; accumulate 0.5*rowsum(W)_i * s_i.
__global__ void he_child(const float* __restrict__ mu, const float* __restrict__ sigma,
                         const float* __restrict__ omega, const float* __restrict__ W,
                         float* __restrict__ X, float* __restrict__ acc) {
  int i = blockIdx.x * blockDim.x + threadIdx.x;
  if (i >= NN) return;

  float O[64], Oi[64], S[64];
#pragma unroll
  for (int t = 0; t < 64; t++) O[t] = omega[i * 64 + t];
#pragma unroll
  for (int t = 0; t < 64; t++) S[t] = sigma[i * 64 + t];
  float ldO = inv8_logdet(O, Oi);
  float ldS = spd_logdet8(S);

  float m[8], u[8];
#pragma unroll
  for (int a = 0; a < 8; a++) m[a] = mu[i * 8 + a];
#pragma unroll
  for (int a = 0; a < 8; a++) {
    float s = 0.0f;
#pragma unroll
    for (int b = 0; b < 8; b++) s += Oi[a * 8 + b] * m[b];
    u[a] = s;
  }

  // tm = Oi * Sigma ; B = tm * Oi^T
  float tm[64];
#pragma unroll
  for (int a = 0; a < 8; a++)
#pragma unroll
    for (int b = 0; b < 8; b++) {
      float s = 0.0f;
#pragma unroll
      for (int c = 0; c < 8; c++) s += Oi[a * 8 + c] * S[c * 8 + b];
      tm[a * 8 + b] = s;
    }

  float* Xr = X + (size_t)i * XC;
#pragma unroll
  for (int a = 0; a < 8; a++)
#pragma unroll
    for (int b = 0; b < 8; b++) {
      float s = 0.0f;
#pragma unroll
      for (int c = 0; c < 8; c++) s += tm[a * 8 + c] * Oi[b * 8 + c];
      Xr[a * 8 + b] = s + u[a] * u[b];
    }
#pragma unroll
  for (int a = 0; a < 8; a++) Xr[64 + a] = -2.0f * u[a];
#pragma unroll
  for (int a = 72; a < XC; a++) Xr[a] = 0.0f;

  float rs = 0.0f;
  for (int c = 0; c < MM; c++) rs += W[i * MM + c];
  atomicAdd(acc, 0.5f * rs * (2.0f * ldO - ldS));
}

// Per parent a: Y_a = [A_flat, w, 0pad]; colsum(W)_a; t_a = c - 2ldO + ldS - K.
__global__ void he_parent(const float* __restrict__ mu, const float* __restrict__ sigma,
                          const float* __restrict__ omega, const float* __restrict__ W,
                          float* __restrict__ Y, float* __restrict__ colsum,
                          float* __restrict__ tcol) {
  int al = threadIdx.x;
  if (al >= MM) return;

  float O[64], Oi[64], S[64], Sinv[64];
#pragma unroll
  for (int t = 0; t < 64; t++) O[t] = omega[al * 64 + t];
#pragma unroll
  for (int t = 0; t < 64; t++) S[t] = sigma[al * 64 + t];
  float ldO = inv8_logdet(O, Oi);
  float ldS = inv8_logdet(S, Sinv);

  // tm = Sinv * O ; A = O^T * tm   (A symmetric)
  float tm[64], A[64];
#pragma unroll
  for (int c = 0; c < 8; c++)
#pragma unroll
    for (int b = 0; b < 8; b++) {
      float s = 0.0f;
#pragma unroll
      for (int d = 0; d < 8; d++) s += Sinv[c * 8 + d] * O[d * 8 + b];
      tm[c * 8 + b] = s;
    }
#pragma unroll
  for (int a = 0; a < 8; a++)
#pragma unroll
    for (int b = 0; b < 8; b++) {
      float s = 0.0f;
#pragma unroll
      for (int c = 0; c < 8; c++) s += O[c * 8 + a] * tm[c * 8 + b];
      A[a * 8 + b] = s;
    }

  float m[8], v[8], w[8];
#pragma unroll
  for (int a = 0; a < 8; a++) m[a] = mu[al * 8 + a];
#pragma unroll
  for (int a = 0; a < 8; a++) {
    float s = 0.0f;
#pragma unroll
    for (int b = 0; b < 8; b++) s += Oi[a * 8 + b] * m[b];
    v[a] = s;
  }
  float cq = 0.0f;
#pragma unroll
  for (int a = 0; a < 8; a++) {
    float s = 0.0f;
#pragma unroll
    for (int b = 0; b < 8; b++) s += A[a * 8 + b] * v[b];
    w[a] = s;
    cq += v[a] * s;
  }

  float* Yr = Y + (size_t)al * XC;
#pragma unroll
  for (int t = 0; t < 64; t++) Yr[t] = A[t];
#pragma unroll
  for (int a = 0; a < 8; a++) Yr[64 + a] = w[a];
#pragma unroll
  for (int a = 72; a < XC; a++) Yr[a] = 0.0f;

  float cs = 0.0f;
  for (int i = 0; i < NN; i++) cs += W[i * MM + al];
  colsum[al] = cs;
  tcol[al] = cq - 2.0f * ldO + ldS - (float)KD;
}

// G[128 x 80] = W^T[128 x 2048] @ X[2048 x 80] via V_WMMA_F32_16X16X4_F32.
// 8 alpha-tiles x 5 feature-tiles = 40 output tiles, one wave each.
// A 16x4 layout: lane 0-15 -> M rows, VGPR0/1 = K {0,1} (lanes<16) / {2,3} (lanes>=16).
// D layout: VGPR v -> M = v + 8*(lane>>4), N = lane&15.
__global__ void he_gemm(const float* __restrict__ W, const float* __restrict__ X,
                        float* __restrict__ G) {
  int wave = (blockIdx.x * blockDim.x + threadIdx.x) >> 5;
  int lane = threadIdx.x & 31;
  int at = wave / 5, ft = wave % 5;
  int abase = at * 16, fbase = ft * 16;
  int row = lane & 15;          // M row (A) / N col (B)
  int kо = (lane >> 4) * 2;     // K sub-offset 0 or 2

  const float* Wp = W + abase + row;   // A[m,k] = W[(k0+k)*MM + abase+m]
  const float* Xp = X + fbase + row;   // B[k,n] = X[(k0+k)*XC + fbase+n]

  v8f c = {};
  for (int k0 = 0; k0 < NN; k0 += 4) {
    v2f a, b;
    a.x = Wp[(k0 + kо) * MM];
    a.y = Wp[(k0 + kо + 1) * MM];
    b.x = Xp[(k0 + kо) * XC];
    b.y = Xp[(k0 + kо + 1) * XC];
    c = __builtin_amdgcn_wmma_f32_16x16x4_f32(false, a, false, b, (short)0, c,
                                              false, false);
  }

  int mbase = abase + (lane >> 4) * 8;
#pragma unroll
  for (int v = 0; v < 8; v++)
    G[(size_t)(mbase + v) * XC + fbase + row] = c[v];
}

// Per parent: 0.5 * ( G[a]·Y[a] + colsum[a]*t[a] ) accumulated into acc[pass].
__global__ void he_final(const float* __restrict__ G, const float* __restrict__ Y,
                         const float* __restrict__ colsum, const float* __restrict__ tcol,
                         float* __restrict__ acc) {
  int al = threadIdx.x;
  if (al >= MM) return;
  float d = 0.0f;
  for (int j = 0; j < 72; j++) d += G[(size_t)al * XC + j] * Y[(size_t)al * XC + j];
  d += colsum[al] * tcol[al];
  atomicAdd(acc, 0.5f * d);
}

__global__ void he_combine(const float* __restrict__ acc, float* __restrict__ out) {
  out[0] = acc[0] + acc[1];  // total
  out[1] = acc[0];           // belief_topdown
  out[2] = acc[1];           // model_topdown
}

// ---------- host launch ----------

extern "C" void kernel_launch(void* const* d_in, const int* in_sizes, int n_in,
                              void* d_out, int out_size, void* d_ws, size_t ws_size,
                              hipStream_t stream) {
  const float* W        = (const float*)d_in[0];
  const float* mu_p     = (const float*)d_in[1];
  const float* sigma_p  = (const float*)d_in[2];
  const float* mu_q     = (const float*)d_in[3];
  const float* sigma_q  = (const float*)d_in[4];
  const float* om_c     = (const float*)d_in[5];
  const float* om_p     = (const float*)d_in[6];
  const float* mu_r     = (const float*)d_in[7];
  const float* sigma_r  = (const float*)d_in[8];
  const float* mu_s     = (const float*)d_in[9];
  const float* sigma_s  = (const float*)d_in[10];
  const float* om_mc    = (const float*)d_in[11];
  const float* om_mp    = (const float*)d_in[12];
  float* out = (float*)d_out;

  float* X      = (float*)d_ws;              // 2048*80
  float* Y      = X + (size_t)NN * XC;       // 128*80
  float* G      = Y + (size_t)MM * XC;       // 128*80
  float* colsum = G + (size_t)MM * XC;       // 128
  float* tcol   = colsum + MM;               // 128
  float* acc    = tcol + MM;                 // 2

  he_init<<<1, 32, 0, stream>>>(acc);

  // pass 0: belief
  he_child<<<NN / 256, 256, 0, stream>>>(mu_p, sigma_p, om_c, W, X, acc + 0);
  he_parent<<<1, MM, 0, stream>>>(mu_q, sigma_q, om_p, W, Y, colsum, tcol);
  he_gemm<<<10, 128, 0, stream>>>(W, X, G);
  he_final<<<1, MM, 0, stream>>>(G, Y, colsum, tcol, acc + 0);

  // pass 1: model (reuses X/Y/G/colsum/tcol — kernels are stream-ordered)
  he_child<<<NN / 256, 256, 0, stream>>>(mu_r, sigma_r, om_mc, W, X, acc + 1);
  he_parent<<<1, MM, 0, stream>>>(mu_s, sigma_s, om_mp, W, Y, colsum, tcol);
  he_gemm<<<10, 128, 0, stream>>>(W, X, G);
  he_final<<<1, MM, 0, stream>>>(G, Y, colsum, tcol, acc + 1);

  he_combine<<<1, 1, 0, stream>>>(acc, out);

  (void)in_sizes; (void)n_in; (void)out_size; (void)ws_size;
}